// Conv2D_87016037416970
// MI455X (gfx1250) — compile-verified
//
#include <hip/hip_runtime.h>

typedef __attribute__((ext_vector_type(2))) float v2f;
typedef __attribute__((ext_vector_type(8))) float v8f;

#define KW     15      // filter size
#define OUT_N  4084    // (4096 + 2*1 - 15) + 1
#define IN_N   4096
#define XS     34      // LDS row stride (floats): even -> 8B-aligned b64, spreads banks
#define NCHUNK 8       // ceil(30/4) chunks of the t (=n+kx) dimension
#define NWAVE  4
#define XROWS  46      // 32 output rows + 15 - 1 halo rows

// Per 16x16 tile: D[m][n] = sum_ky sum_t X[m+ky][t] * w[ky][t-n]
// via V_WMMA_F32_16X16X4_F32. Each wave does TWO vertically stacked tiles
// (independent accumulator chains, shared B operands). Fully unrolled so the
// 240 WMMAs form a straight-line stream with no accumulator copies.
__global__ __launch_bounds__(128) void conv15_wmma_f32(const float* __restrict__ x,
                                                       const float* __restrict__ w,
                                                       const float* __restrict__ bias,
                                                       float* __restrict__ out)
{
    // Banded-Toeplitz weight tiles: B[ky][j][k][n] = w[ky][4j+k-n] (0 outside band).
    // Identical for every output tile -> built once per workgroup.
    __shared__ __align__(16) float Btab[KW * NCHUNK * 32 * 2];   // 30720 B
    __shared__ __align__(16) float Xt[NWAVE][XROWS * XS];        // 4 * 6256 B

    const int tid  = threadIdx.x;
    const int lane = tid & 31;
    const int wv   = tid >> 5;

    // ---- stage banded B tiles (distributed over all 128 threads) ----
    // entry e = (ky*8 + j)*32 + virtual_lane ; one float2 per entry
    for (int e = tid; e < KW * NCHUNK * 32; e += 128) {
        int vl = e & 31;
        int pj = e >> 5;
        int ky = pj >> 3;          // pj / 8
        int j  = pj & 7;
        int kh = (vl >> 4) << 1;   // hi lanes hold K=2,3
        int n  = vl & 15;
        int d0 = 4 * j + kh - n;
        int d1 = d0 + 1;
        float b0 = ((unsigned)d0 < (unsigned)KW) ? w[ky * KW + d0] : 0.0f;
        float b1 = ((unsigned)d1 < (unsigned)KW) ? w[ky * KW + d1] : 0.0f;
        Btab[e * 2 + 0] = b0;
        Btab[e * 2 + 1] = b1;
    }

    // ---- stage per-wave 46x32 zero-padded input halo ----
    // Wave strip: 32 output rows x 16 output cols.
    const int row0 = blockIdx.y * 64 + (wv >> 1) * 32;  // output-space strip origin
    const int col0 = blockIdx.x * 32 + (wv & 1) * 16;
    {
        int  ic  = col0 - 1 + lane;                      // PAD = 1
        bool cok = (unsigned)ic < (unsigned)IN_N;
        float* xt = &Xt[wv][0];
        for (int r = 0; r < XROWS; ++r) {
            int ir = row0 - 1 + r;
            float v = 0.0f;
            if (cok && (unsigned)ir < (unsigned)IN_N) v = x[ir * IN_N + ic];
            xt[r * XS + lane] = v;
        }
    }
    __syncthreads();

    // ---- WMMA accumulation: two independent accumulator chains ----
    const int mA = lane & 15;           // A: lane holds row M = lane&15
    const int kh = (lane >> 4) << 1;    // A/B: lanes 16-31 hold K=2,3
    const float bv = bias[0];
    v8f c0 = { bv, bv, bv, bv, bv, bv, bv, bv };   // rows row0 .. row0+15
    v8f c1 = c0;                                   // rows row0+16 .. row0+31

    const float* xt = &Xt[wv][0];
#pragma unroll
    for (int ky = 0; ky < KW; ++ky) {
        const float* arow0 = &xt[(mA + ky) * XS + kh];
        const float* arow1 = arow0 + 16 * XS;
        const float* brow  = &Btab[ky * (NCHUNK * 64) + lane * 2];
        v2f bt[NCHUNK];
#pragma unroll
        for (int j = 0; j < NCHUNK; ++j)
            bt[j] = *(const v2f*)(brow + j * 64);
#pragma unroll
        for (int j = 0; j < NCHUNK; ++j) {
            v2f a0 = *(const v2f*)(arow0 + 4 * j);
            v2f a1 = *(const v2f*)(arow1 + 4 * j);
            c0 = __builtin_amdgcn_wmma_f32_16x16x4_f32(
                     false, a0, false, bt[j], (short)0, c0, false, false);
            c1 = __builtin_amdgcn_wmma_f32_16x16x4_f32(
                     false, a1, false, bt[j], (short)0, c1, false, false);
        }
    }

    // ---- store: C/D layout -> VGPR v holds M = v + 8*hi, N = lane&15 ----
    const int n  = lane & 15;
    const int hi = lane >> 4;
    const int oc = col0 + n;
    if (oc < OUT_N) {
#pragma unroll
        for (int v = 0; v < 8; ++v) {
            int om0 = row0 + v + 8 * hi;
            int om1 = om0 + 16;
            if (om0 < OUT_N) out[om0 * OUT_N + oc] = c0[v];
            if (om1 < OUT_N) out[om1 * OUT_N + oc] = c1[v];
        }
    }
}

extern "C" void kernel_launch(void* const* d_in, const int* in_sizes, int n_in,
                              void* d_out, int out_size, void* d_ws, size_t ws_size,
                              hipStream_t stream) {
    const float* x    = (const float*)d_in[0];  // 4096*4096
    const float* wgt  = (const float*)d_in[1];  // 15*15
    const float* bias = (const float*)d_in[2];  // 1
    float* out = (float*)d_out;                 // 4084*4084

    dim3 grid((OUT_N + 31) / 32, (OUT_N + 63) / 64);
    conv15_wmma_f32<<<grid, 128, 0, stream>>>(x, wgt, bias, out);
}